// BaseLTI_31490700215027
// MI455X (gfx1250) — compile-verified
//
#include <hip/hip_runtime.h>

// Kalman filter (continuous-discrete LTI) for MI455X gfx1250.
// One wave32 per batch element; all dense 32/16-dim GEMMs via
// V_WMMA_F32_16X16X4_F32 with operands staged in padded LDS.

#ifdef __HIP_DEVICE_COMPILE__
#if !__has_builtin(__builtin_amdgcn_wmma_f32_16x16x4_f32)
#error "wmma f32 16x16x4 builtin missing on this toolchain"
#endif
#endif

typedef float v2f __attribute__((ext_vector_type(2)));
typedef float v8f __attribute__((ext_vector_type(8)));

#define ZD 32
#define YD 16
#define LDZ 33   // padded leading dim for 32-col matrices
#define LDY 17   // padded leading dim for 16-col matrices

// C(MxN) (+)= A(MxK) * B(KxN)   [row-major LDS, TRANSB => B given as NxK]
// WMMA f32 16x16x4 fragment layout (wave32):
//   A: lane l -> row l%16, K = k0 + 2*(l/16) + {0,1}
//   B: lane l -> col l%16, K = k0 + 2*(l/16) + {0,1}
//   C/D: vgpr v, lane l -> row v + 8*(l/16), col l%16
template <int M, int N, int K, bool TRANSB, bool ACCUM>
__device__ __forceinline__ void mm_tiles(float* D, int ldd,
                                         const float* A, int lda,
                                         const float* Bm, int ldb,
                                         int lane) {
  const int m = lane & 15;
  const int g = lane >> 4;
  for (int mi = 0; mi < M; mi += 16) {
    for (int ni = 0; ni < N; ni += 16) {
      v8f acc;
#pragma unroll
      for (int v = 0; v < 8; ++v)
        acc[v] = ACCUM ? D[(mi + v + 8 * g) * ldd + (ni + m)] : 0.0f;
#pragma unroll
      for (int k0 = 0; k0 < K; k0 += 4) {
        const int ka = k0 + 2 * g;
        v2f a, b;
        a[0] = A[(mi + m) * lda + ka + 0];
        a[1] = A[(mi + m) * lda + ka + 1];
        if (TRANSB) {
          b[0] = Bm[(ni + m) * ldb + ka + 0];
          b[1] = Bm[(ni + m) * ldb + ka + 1];
        } else {
          b[0] = Bm[(ka + 0) * ldb + (ni + m)];
          b[1] = Bm[(ka + 1) * ldb + (ni + m)];
        }
        acc = __builtin_amdgcn_wmma_f32_16x16x4_f32(
            false, a, false, b, (short)0, acc, false, false);
      }
#pragma unroll
      for (int v = 0; v < 8; ++v)
        D[(mi + v + 8 * g) * ldd + (ni + m)] = acc[v];
    }
  }
}

// In-place cooperative Cholesky (lower) of n x n SPD matrix in LDS.
// Lane i owns row i. Upper triangle left stale (caller zeroes if needed).
__device__ __forceinline__ void cholesky_ip(float* S, int n, int ld, int lane) {
  for (int k = 0; k < n; ++k) {
    __syncthreads();
    float dk = sqrtf(S[k * ld + k]);
    __syncthreads();
    if (lane == k) S[k * ld + k] = dk;
    if (lane > k && lane < n) S[lane * ld + k] /= dk;
    __syncthreads();
    if (lane > k && lane < n) {
      float lik = S[lane * ld + k];
      for (int j = k + 1; j <= lane; ++j)
        S[lane * ld + j] -= lik * S[j * ld + k];
    }
  }
  __syncthreads();
}

__global__ __launch_bounds__(32) void kf_lti_kernel(
    const float* __restrict__ y, const float* __restrict__ mask,
    const float* __restrict__ Fg, const float* __restrict__ Hg,
    const float* __restrict__ qdiag, const float* __restrict__ rdiag,
    const float* __restrict__ mu0, const float* __restrict__ sig0diag,
    float* __restrict__ out_mus, float* __restrict__ out_Ls,
    float* __restrict__ out_logp, int B, int T) {
  const int b = blockIdx.x;
  const int lane = threadIdx.x;
  const float STEP = 0.05f;
  const float LOG2PI = 1.83787706640934548356f;

  __shared__ float sF[ZD * LDZ];    // F (32x32)
  __shared__ float sH[YD * LDZ];    // H (16x32)
  __shared__ float sSIG[ZD * LDZ];  // Sigma scratch
  __shared__ float sLp[ZD * LDZ];   // carried Cholesky factor L_pred
  __shared__ float sLu[ZD * LDZ];   // updated Cholesky factor
  __shared__ float sA[ZD * LDZ];    // A = I - K H   (also Euler T1)
  __shared__ float sM2[ZD * LDZ];   // A*Sigma       (also Euler T2)
  __shared__ float sHS[YD * LDZ];   // H * Sigma (16x32)
  __shared__ float sS[YD * LDY];    // innovation cov / its chol (16x16)
  __shared__ float sK[ZD * LDY];    // Kalman gain (32x16)
  __shared__ float sKR[ZD * LDY];   // K * diag(R)  (32x16)
  __shared__ float sQ[ZD], sR[YD], smu[ZD], smu2[ZD], sInn[YD];

  // ---- load constants / init carry ----
  for (int i = lane; i < ZD * ZD; i += 32) {
    int r = i >> 5, c = i & 31;
    sF[r * LDZ + c] = Fg[i];
    sLp[r * LDZ + c] = 0.0f;
  }
  for (int i = lane; i < YD * ZD; i += 32) {
    int r = i >> 5, c = i & 31;
    sH[r * LDZ + c] = Hg[i];
  }
  sQ[lane] = qdiag[lane];
  smu[lane] = mu0[lane];
  if (lane < YD) sR[lane] = rdiag[lane];
  __syncthreads();
  sLp[lane * LDZ + lane] = sqrtf(sig0diag[lane]);  // L_pred0 = diag(sqrt(sig0))
  __syncthreads();

  float logp_acc = 0.0f;

  for (int t = 0; t < T; ++t) {
    // 1. Sigma_pred = Lp * Lp^T
    mm_tiles<ZD, ZD, ZD, true, false>(sSIG, LDZ, sLp, LDZ, sLp, LDZ, lane);
    __syncthreads();
    // 2. HS = H * Sigma_pred (16x32)
    mm_tiles<YD, ZD, ZD, false, false>(sHS, LDZ, sH, LDZ, sSIG, LDZ, lane);
    __syncthreads();
    // 3. S = HS * H^T + R
    mm_tiles<YD, YD, ZD, true, false>(sS, LDY, sHS, LDZ, sH, LDZ, lane);
    __syncthreads();
    if (lane < YD) sS[lane * LDY + lane] += sR[lane];
    __syncthreads();
    // 4. LS = chol(S) (lower, in place)
    cholesky_ip(sS, YD, LDY, lane);
    // 5. innovation = y - H mu_pred
    if (lane < YD) {
      float yh = 0.0f;
      for (int z = 0; z < ZD; ++z) yh += sH[lane * LDZ + z] * smu[z];
      sInn[lane] = y[((size_t)b * T + t) * YD + lane] - yh;
    }
    __syncthreads();
    // 6. K: solve S * K^T = PHt^T, lane c owns column c (PHt[c][i] = HS[i][c])
    {
      const int c = lane;
      float x[YD];
#pragma unroll
      for (int i = 0; i < YD; ++i) {
        float s = sHS[i * LDZ + c];
        for (int j = 0; j < i; ++j) s -= sS[i * LDY + j] * x[j];
        x[i] = s / sS[i * LDY + i];
      }
#pragma unroll
      for (int i = YD - 1; i >= 0; --i) {
        float s = x[i];
        for (int j = i + 1; j < YD; ++j) s -= sS[j * LDY + i] * x[j];
        x[i] = s / sS[i * LDY + i];
      }
#pragma unroll
      for (int i = 0; i < YD; ++i) sK[c * LDY + i] = x[i];  // K[c][i]
    }
    __syncthreads();
    // 7. mu_u = mu_pred + K * innov (lane = z)
    float mu_u_l = smu[lane];
    for (int yi = 0; yi < YD; ++yi) mu_u_l += sK[lane * LDY + yi] * sInn[yi];
    // 8. A = I - K*H
    mm_tiles<ZD, ZD, YD, false, false>(sA, LDZ, sK, LDY, sH, LDZ, lane);
    __syncthreads();
    for (int i = lane; i < ZD * ZD; i += 32) {
      int r = i >> 5, c = i & 31;
      sA[r * LDZ + c] = (r == c ? 1.0f : 0.0f) - sA[r * LDZ + c];
    }
    __syncthreads();
    // 9-11. Sigma_u = A Sig A^T + (K R) K^T  (Joseph form)
    mm_tiles<ZD, ZD, ZD, false, false>(sM2, LDZ, sA, LDZ, sSIG, LDZ, lane);
    __syncthreads();
    mm_tiles<ZD, ZD, ZD, true, false>(sSIG, LDZ, sM2, LDZ, sA, LDZ, lane);
    __syncthreads();
    for (int i = lane; i < ZD * YD; i += 32) {
      int r = i >> 4, c = i & 15;
      sKR[r * LDY + c] = sK[r * LDY + c] * sR[c];
    }
    __syncthreads();
    mm_tiles<ZD, ZD, YD, true, true>(sSIG, LDZ, sKR, LDY, sK, LDY, lane);
    __syncthreads();
    // 12. Lu = chol(0.5*(Sig_u + Sig_u^T))
    for (int i = lane; i < ZD * ZD; i += 32) {
      int r = i >> 5, c = i & 31;
      sLu[r * LDZ + c] = 0.5f * (sSIG[r * LDZ + c] + sSIG[c * LDZ + r]);
    }
    __syncthreads();
    cholesky_ip(sLu, ZD, LDZ, lane);
    for (int i = lane; i < ZD * ZD; i += 32) {
      int r = i >> 5, c = i & 31;
      if (c > r) sLu[r * LDZ + c] = 0.0f;
    }
    __syncthreads();
    // 13. log-prob of innovation under N(0, S) (redundant across lanes)
    const float m_i = mask[(size_t)b * T + t];
    {
      float sol[YD];
      float sumlog = 0.0f, qq = 0.0f;
#pragma unroll
      for (int i = 0; i < YD; ++i) {
        float s = sInn[i];
        for (int j = 0; j < i; ++j) s -= sS[i * LDY + j] * sol[j];
        sol[i] = s / sS[i * LDY + i];
        sumlog += logf(sS[i * LDY + i]);
        qq += sol[i] * sol[i];
      }
      logp_acc += m_i * (-0.5f * (float)YD * LOG2PI - sumlog - 0.5f * qq);
    }
    // 14-15. mask-blend, emit outputs, update carry L
    {
      float mu_b = m_i * mu_u_l + (1.0f - m_i) * smu[lane];
      smu2[lane] = mu_b;
      out_mus[((size_t)t * B + b) * ZD + lane] = mu_b;
    }
    for (int i = lane; i < ZD * ZD; i += 32) {
      int r = i >> 5, c = i & 31;
      float lb = m_i * sLu[r * LDZ + c] + (1.0f - m_i) * sLp[r * LDZ + c];
      sLp[r * LDZ + c] = lb;
      out_Ls[(((size_t)t * B + b) << 10) + i] = lb;
    }
    __syncthreads();
    smu[lane] = smu2[lane];
    __syncthreads();
    // 16. Sigma = L * L^T (blended L)
    mm_tiles<ZD, ZD, ZD, true, false>(sSIG, LDZ, sLp, LDZ, sLp, LDZ, lane);
    __syncthreads();
    // 17. two Euler steps of moment ODEs
    for (int e = 0; e < 2; ++e) {
      {
        float s = 0.0f;
        for (int k = 0; k < ZD; ++k) s += sF[lane * LDZ + k] * smu[k];
        smu2[lane] = smu[lane] + STEP * s;
      }
      mm_tiles<ZD, ZD, ZD, false, false>(sA, LDZ, sF, LDZ, sSIG, LDZ, lane);
      __syncthreads();
      mm_tiles<ZD, ZD, ZD, true, false>(sM2, LDZ, sSIG, LDZ, sF, LDZ, lane);
      __syncthreads();
      for (int i = lane; i < ZD * ZD; i += 32) {
        int r = i >> 5, c = i & 31;
        sSIG[r * LDZ + c] += STEP * (sA[r * LDZ + c] + sM2[r * LDZ + c] +
                                     (r == c ? sQ[r] : 0.0f));
      }
      smu[lane] = smu2[lane];
      __syncthreads();
    }
    // 18. L_pred(next) = chol(0.5*(Sig + Sig^T))
    for (int i = lane; i < ZD * ZD; i += 32) {
      int r = i >> 5, c = i & 31;
      sLp[r * LDZ + c] = 0.5f * (sSIG[r * LDZ + c] + sSIG[c * LDZ + r]);
    }
    __syncthreads();
    cholesky_ip(sLp, ZD, LDZ, lane);
    for (int i = lane; i < ZD * ZD; i += 32) {
      int r = i >> 5, c = i & 31;
      if (c > r) sLp[r * LDZ + c] = 0.0f;
    }
    __syncthreads();
  }

  if (lane == 0) out_logp[b] = logp_acc;
}

extern "C" void kernel_launch(void* const* d_in, const int* in_sizes, int n_in,
                              void* d_out, int out_size, void* d_ws,
                              size_t ws_size, hipStream_t stream) {
  const float* y = (const float*)d_in[0];
  const float* mask = (const float*)d_in[1];
  // d_in[2] = times (uniform spacing, unused)
  const float* F = (const float*)d_in[3];
  const float* H = (const float*)d_in[4];
  const float* qdiag = (const float*)d_in[5];
  const float* rdiag = (const float*)d_in[6];
  const float* mu0 = (const float*)d_in[7];
  const float* sig0 = (const float*)d_in[8];

  const int T = in_sizes[2];           // 128
  const int B = in_sizes[1] / T;       // 512

  float* out = (float*)d_out;
  float* out_mus = out;                                  // (T,B,32)
  float* out_Ls = out + (size_t)T * B * ZD;              // (T,B,32,32)
  float* out_logp = out_Ls + (size_t)T * B * ZD * ZD;    // (B,)

  hipLaunchKernelGGL(kf_lti_kernel, dim3(B), dim3(32), 0, stream, y, mask, F,
                     H, qdiag, rdiag, mu0, sig0, out_mus, out_Ls, out_logp, B,
                     T);
}